// IsotropicImplementation_11098195493542
// MI455X (gfx1250) — compile-verified
//
#include <hip/hip_runtime.h>
#include <hip/hip_bf16.h>

// Problem constants (match reference setup_inputs)
constexpr int N = 6;      // state rows
constexpr int D = 1024;   // state cols
constexpr int T = 4096;   // time steps
constexpr int C = 128;    // chunks
constexpr int L = 32;     // steps per chunk (T = C*L)
constexpr int P = 48;     // padded Kronecker dim (36 -> 48, 3x3 WMMA tiles)
static_assert(T == C * L, "chunking");

typedef __attribute__((ext_vector_type(2))) float v2f;
typedef __attribute__((ext_vector_type(8))) float v8f;

// ---------------------------------------------------------------------------
// Kernel 1: per-chunk local means scan with zero seed -> chunk offset u_c
// grid = C, block = 256 (each thread owns 4 columns)
// ---------------------------------------------------------------------------
__global__ void k_means_local(const float* __restrict__ trans,
                              const float* __restrict__ nmean,
                              float* __restrict__ u) {
  __shared__ float sA[L * 36];
  const int c = blockIdx.x;
  const int tid = threadIdx.x;
  for (int idx = tid; idx < L * 36; idx += 256)
    sA[idx] = trans[(long)c * L * 36 + idx];
  __syncthreads();

  const int j0 = tid * 4;
  float s[N][4];
#pragma unroll
  for (int r = 0; r < N; ++r)
#pragma unroll
    for (int x = 0; x < 4; ++x) s[r][x] = 0.f;

  for (int k = 0; k < L; ++k) {
    const long t = (long)c * L + k;
    const float* Ak = &sA[k * 36];
    float ns[N][4];
#pragma unroll
    for (int r = 0; r < N; ++r) {
      const float4 b = *(const float4*)&nmean[(t * N + r) * D + j0];
      float ax = b.x, ay = b.y, az = b.z, aw = b.w;
#pragma unroll
      for (int m = 0; m < N; ++m) {
        const float a = Ak[r * N + m];
        ax += a * s[m][0]; ay += a * s[m][1]; az += a * s[m][2]; aw += a * s[m][3];
      }
      ns[r][0] = ax; ns[r][1] = ay; ns[r][2] = az; ns[r][3] = aw;
    }
#pragma unroll
    for (int r = 0; r < N; ++r)
#pragma unroll
      for (int x = 0; x < 4; ++x) s[r][x] = ns[r][x];
  }
#pragma unroll
  for (int r = 0; r < N; ++r) {
    float4 v; v.x = s[r][0]; v.y = s[r][1]; v.z = s[r][2]; v.w = s[r][3];
    *(float4*)&u[((long)c * N + r) * D + j0] = v;
  }
}

// ---------------------------------------------------------------------------
// Kernel 2: chunk 6x6 matrix products G_c = A_{last} ... A_{first}
// grid = 1, block = C (one thread per chunk, serial 6x6 matmuls)
// ---------------------------------------------------------------------------
__global__ void k_products(const float* __restrict__ trans,
                           float* __restrict__ G) {
  const int c = threadIdx.x;  // blockDim.x == C
  float g[36], gn[36], a[36];
#pragma unroll
  for (int i = 0; i < 36; ++i) g[i] = (i % 7 == 0) ? 1.f : 0.f;  // I6
  for (int k = 0; k < L; ++k) {
    const float* A = &trans[((long)c * L + k) * 36];
#pragma unroll
    for (int i = 0; i < 36; ++i) a[i] = A[i];
#pragma unroll
    for (int r = 0; r < 6; ++r)
#pragma unroll
      for (int cc = 0; cc < 6; ++cc) {
        float acc = 0.f;
#pragma unroll
        for (int m = 0; m < 6; ++m) acc += a[r * 6 + m] * g[m * 6 + cc];
        gn[r * 6 + cc] = acc;
      }
#pragma unroll
    for (int i = 0; i < 36; ++i) g[i] = gn[i];
  }
#pragma unroll
  for (int i = 0; i < 36; ++i) G[c * 36 + i] = g[i];
}

// ---------------------------------------------------------------------------
// Kernel 3: sequential chunk-level means combine -> chunk-start seeds
// grid = 1, block = D (thread j owns column j)
// ---------------------------------------------------------------------------
__global__ void k_means_seq(const float* __restrict__ init_mean,
                            const float* __restrict__ G,
                            const float* __restrict__ u,
                            float* __restrict__ seeds) {
  const int j = threadIdx.x;  // 0..1023
  float s[N];
#pragma unroll
  for (int r = 0; r < N; ++r) s[r] = init_mean[r * D + j];
  for (int c = 0; c < C; ++c) {
#pragma unroll
    for (int r = 0; r < N; ++r) seeds[((long)c * N + r) * D + j] = s[r];
    const float* g = &G[c * 36];  // uniform -> scalar loads
    float ns[N];
#pragma unroll
    for (int r = 0; r < N; ++r) {
      float acc = u[((long)c * N + r) * D + j];
#pragma unroll
      for (int m = 0; m < N; ++m) acc += g[r * N + m] * s[m];
      ns[r] = acc;
    }
#pragma unroll
    for (int r = 0; r < N; ++r) s[r] = ns[r];
  }
}

// ---------------------------------------------------------------------------
// Kernel 4: per-chunk Kronecker products M_c = prod (A_t (x) A_t)  [48x48 pad]
// via V_WMMA_F32_16X16X4_F32 (9 waves = 3x3 tiles of 16x16), plus offsets q_c.
// K = A (x) A has zero columns for j>=36, so the k-loop runs only kk<36
// (9 WMMAs per step). A-operand generated branchlessly in registers.
// grid = C, block = 288 (9 waves)
// ---------------------------------------------------------------------------
__global__ void k_cov_kron(const float* __restrict__ trans,
                           const float* __restrict__ ncov,
                           float* __restrict__ Mws,
                           float* __restrict__ qws) {
  __shared__ float Mb[P * P];
  __shared__ float sA[36];
  __shared__ float sNL[36];
  __shared__ float sq[36];
  const int c = blockIdx.x;
  const int tid = threadIdx.x;   // 0..287
  const int lane = tid & 31;
  const int wave = tid >> 5;     // 0..8
  const int ti = wave / 3, tj = wave % 3;
  const int rb = ti * 16, cb = tj * 16;
  const int lrow = rb + (lane & 15);  // A-operand row for this lane
  const float rmask = (lrow < 36) ? 1.f : 0.f;
  const int ldiv = (lrow < 36) ? (lrow / 6) : 0;
  const int lmod = (lrow < 36) ? (lrow % 6) : 0;
  const bool lo16 = (lane < 16);
  const int cOff = lo16 ? 0 : 2;
  const int mcol = cb + (lane & 15);

  for (int idx = tid; idx < P * P; idx += 288)
    Mb[idx] = ((idx / P) == (idx % P)) ? 1.f : 0.f;
  if (tid < 36) sq[tid] = 0.f;
  __syncthreads();

  for (int k = 0; k < L; ++k) {
    const long t = (long)c * L + k;
    if (tid < 36) sA[tid] = trans[t * 36 + tid];
    else if (tid >= 64 && tid < 100) sNL[tid - 64] = ncov[t * 36 + (tid - 64)];
    __syncthreads();

    // Per-lane row factors of K in registers (row mask folded in).
    float ah[6], al[6];
#pragma unroll
    for (int m = 0; m < 6; ++m) {
      ah[m] = sA[ldiv * 6 + m] * rmask;
      al[m] = sA[lmod * 6 + m];
    }

    // D_tile = K(48x36) * Mb(36xP), tile (ti,tj); K[r][j] = A[r/6][j/6]*A[r%6][j%6]
    v8f acc = {0.f, 0.f, 0.f, 0.f, 0.f, 0.f, 0.f, 0.f};
#pragma unroll
    for (int kk = 0; kk < 36; kk += 4) {
      // compile-time column indices: lanes 0-15 use kk,kk+1; lanes 16-31 kk+2,kk+3
      const float pa0 = ah[(kk + 0) / 6] * al[(kk + 0) % 6];
      const float pa1 = ah[(kk + 1) / 6] * al[(kk + 1) % 6];
      const float pb0 = ah[(kk + 2) / 6] * al[(kk + 2) % 6];
      const float pb1 = ah[(kk + 3) / 6] * al[(kk + 3) % 6];
      v2f av; av.x = lo16 ? pa0 : pb0; av.y = lo16 ? pa1 : pb1;
      const int c0 = kk + cOff;
      v2f bv; bv.x = Mb[c0 * P + mcol]; bv.y = Mb[(c0 + 1) * P + mcol];
      acc = __builtin_amdgcn_wmma_f32_16x16x4_f32(
          false, av, false, bv, (short)0, acc, false, false);
    }

    // q <- K q + vec(n_l n_l^T)   (threads 0..35; reads old sq before barrier)
    float qn = 0.f;
    if (tid < 36) {
      const int a_ = tid / 6, b_ = tid % 6;
      float s0 = 0.f;
#pragma unroll
      for (int m = 0; m < 6; ++m) s0 += sNL[a_ * 6 + m] * sNL[b_ * 6 + m];
      float acq = s0;
      for (int j = 0; j < 36; ++j)
        acq += sA[a_ * 6 + j / 6] * sA[b_ * 6 + j % 6] * sq[j];
      qn = acq;
    }
    __syncthreads();  // all reads of Mb/sq done -> safe in-place write

    const int rbase = rb + (lo16 ? 0 : 8);
#pragma unroll
    for (int v = 0; v < 8; ++v) Mb[(rbase + v) * P + mcol] = acc[v];
    if (tid < 36) sq[tid] = qn;
    __syncthreads();
  }

  for (int idx = tid; idx < P * P; idx += 288)
    Mws[(long)c * P * P + idx] = Mb[idx];
  if (tid < 36) qws[c * 36 + tid] = sq[tid];
}

// ---------------------------------------------------------------------------
// Kernel 5: sequential chunk-level cov combine -> chunk-start vec(Sigma) seeds
// grid = 1, block = 64
// ---------------------------------------------------------------------------
__global__ void k_cov_seq(const float* __restrict__ init_cov,
                          const float* __restrict__ Mws,
                          const float* __restrict__ qws,
                          float* __restrict__ cs) {
  __shared__ float s[36], sn[36];
  const int tid = threadIdx.x;
  if (tid < 36) {
    const int a_ = tid / 6, b_ = tid % 6;
    float acc = 0.f;
#pragma unroll
    for (int m = 0; m < 6; ++m) acc += init_cov[a_ * 6 + m] * init_cov[b_ * 6 + m];
    s[tid] = acc;
  }
  __syncthreads();
  for (int c = 0; c < C; ++c) {
    if (tid < 36) {
      cs[c * 36 + tid] = s[tid];
      const float* M = &Mws[(long)c * P * P + tid * P];
      float acc = qws[c * 36 + tid];
      for (int j = 0; j < 36; ++j) acc += M[j] * s[j];
      sn[tid] = acc;
    }
    __syncthreads();
    if (tid < 36) s[tid] = sn[tid];
    __syncthreads();
  }
}

// ---------------------------------------------------------------------------
// Kernel 6: per-chunk seeded means re-scan, writing all outputs
// grid = C, block = 256
// ---------------------------------------------------------------------------
__global__ void k_means_final(const float* __restrict__ trans,
                              const float* __restrict__ nmean,
                              const float* __restrict__ seeds,
                              float* __restrict__ means_out) {
  __shared__ float sA[L * 36];
  const int c = blockIdx.x;
  const int tid = threadIdx.x;
  for (int idx = tid; idx < L * 36; idx += 256)
    sA[idx] = trans[(long)c * L * 36 + idx];
  __syncthreads();

  const int j0 = tid * 4;
  float s[N][4];
#pragma unroll
  for (int r = 0; r < N; ++r) {
    float4 v = *(const float4*)&seeds[((long)c * N + r) * D + j0];
    s[r][0] = v.x; s[r][1] = v.y; s[r][2] = v.z; s[r][3] = v.w;
  }
  for (int k = 0; k < L; ++k) {
    const long t = (long)c * L + k;
    const float* Ak = &sA[k * 36];
    float ns[N][4];
#pragma unroll
    for (int r = 0; r < N; ++r) {
      const float4 b = *(const float4*)&nmean[(t * N + r) * D + j0];
      float ax = b.x, ay = b.y, az = b.z, aw = b.w;
#pragma unroll
      for (int m = 0; m < N; ++m) {
        const float a = Ak[r * N + m];
        ax += a * s[m][0]; ay += a * s[m][1]; az += a * s[m][2]; aw += a * s[m][3];
      }
      float4 o; o.x = ax; o.y = ay; o.z = az; o.w = aw;
      *(float4*)&means_out[(t * N + r) * D + j0] = o;
      ns[r][0] = ax; ns[r][1] = ay; ns[r][2] = az; ns[r][3] = aw;
    }
#pragma unroll
    for (int r = 0; r < N; ++r)
#pragma unroll
      for (int x = 0; x < 4; ++x) s[r][x] = ns[r][x];
  }
}

// ---------------------------------------------------------------------------
// Kernel 7: per-chunk seeded Sigma re-scan + per-step Cholesky -> covs output
// grid = C, block = 64
// ---------------------------------------------------------------------------
__global__ void k_cov_final(const float* __restrict__ trans,
                            const float* __restrict__ ncov,
                            const float* __restrict__ cs,
                            float* __restrict__ covs_out) {
  __shared__ float Sig[36], Bm[36], Sn[36], sA[36], sNL[36], Lo[36];
  const int c = blockIdx.x;
  const int tid = threadIdx.x;
  if (tid < 36) Sig[tid] = cs[c * 36 + tid];
  for (int k = 0; k < L; ++k) {
    const long t = (long)c * L + k;
    if (tid < 36) { sA[tid] = trans[t * 36 + tid]; sNL[tid] = ncov[t * 36 + tid]; }
    __syncthreads();
    if (tid < 36) {  // B = A * Sigma
      const int a_ = tid / 6, b_ = tid % 6;
      float acc = 0.f;
#pragma unroll
      for (int m = 0; m < 6; ++m) acc += sA[a_ * 6 + m] * Sig[m * 6 + b_];
      Bm[tid] = acc;
    }
    __syncthreads();
    if (tid < 36) {  // Sn = B * A^T + NL * NL^T
      const int a_ = tid / 6, b_ = tid % 6;
      float acc = 0.f;
#pragma unroll
      for (int m = 0; m < 6; ++m) acc += Bm[a_ * 6 + m] * sA[b_ * 6 + m];
#pragma unroll
      for (int m = 0; m < 6; ++m) acc += sNL[a_ * 6 + m] * sNL[b_ * 6 + m];
      Sn[tid] = acc;
    }
    __syncthreads();
    if (tid < 36) Sig[tid] = Sn[tid];
    __syncthreads();
    if (tid == 0) {  // lower Cholesky of Sig -> Lo
      for (int j = 0; j < 6; ++j) {
        float d = Sig[j * 6 + j];
        for (int m = 0; m < j; ++m) d -= Lo[j * 6 + m] * Lo[j * 6 + m];
        d = d > 0.f ? sqrtf(d) : 0.f;
        Lo[j * 6 + j] = d;
        const float inv = d > 0.f ? 1.f / d : 0.f;
        for (int i = j + 1; i < 6; ++i) {
          float acc = Sig[i * 6 + j];
          for (int m = 0; m < j; ++m) acc -= Lo[i * 6 + m] * Lo[j * 6 + m];
          Lo[i * 6 + j] = acc * inv;
        }
        for (int i = 0; i < j; ++i) Lo[i * 6 + j] = 0.f;
      }
    }
    __syncthreads();
    if (tid < 36) covs_out[t * 36 + tid] = Lo[tid];
    __syncthreads();
  }
}

// ---------------------------------------------------------------------------
extern "C" void kernel_launch(void* const* d_in, const int* in_sizes, int n_in,
                              void* d_out, int out_size, void* d_ws, size_t ws_size,
                              hipStream_t stream) {
  (void)in_sizes; (void)n_in; (void)out_size; (void)ws_size;
  const float* init_mean = (const float*)d_in[0];  // [6,1024]
  const float* init_cov  = (const float*)d_in[1];  // [6,6]
  const float* trans     = (const float*)d_in[2];  // [T,6,6]
  const float* nmean     = (const float*)d_in[3];  // [T,6,1024]
  const float* ncov      = (const float*)d_in[4];  // [T,6,6]

  float* means_out = (float*)d_out;                       // [T,6,1024]
  float* covs_out  = means_out + (long)T * N * D;         // [T,6,6]

  // workspace layout (floats): ~7.2 MB total
  float* ws    = (float*)d_ws;
  float* u     = ws;                   // C*N*D
  float* seeds = u + (long)C * N * D;  // C*N*D
  float* G     = seeds + (long)C * N * D;  // C*36
  float* Mws   = G + (long)C * 36;         // C*P*P
  float* qws   = Mws + (long)C * P * P;    // C*36
  float* cseed = qws + (long)C * 36;       // C*36

  // means pipeline
  k_means_local<<<C, 256, 0, stream>>>(trans, nmean, u);
  k_products<<<1, C, 0, stream>>>(trans, G);
  k_means_seq<<<1, D, 0, stream>>>(init_mean, G, u, seeds);
  k_means_final<<<C, 256, 0, stream>>>(trans, nmean, seeds, means_out);

  // covariance pipeline (Kronecker-lifted parallel scan, WMMA chunk products)
  k_cov_kron<<<C, 288, 0, stream>>>(trans, ncov, Mws, qws);
  k_cov_seq<<<1, 64, 0, stream>>>(init_cov, Mws, qws, cseed);
  k_cov_final<<<C, 64, 0, stream>>>(trans, ncov, cseed, covs_out);
}